// MeshGraphUnet2_90400471646659
// MI455X (gfx1250) — compile-verified
//
#include <hip/hip_runtime.h>
#include <math.h>

typedef __attribute__((ext_vector_type(2))) float v2f;
typedef __attribute__((ext_vector_type(8))) float v8f;

#define WAVE 32

__device__ __forceinline__ float waveReduceSum(float v) {
  #pragma unroll
  for (int off = 16; off > 0; off >>= 1) v += __shfl_xor(v, off, WAVE);
  return v;
}

// ---------------------------------------------------------------------------
// score[i] = tanh( dot(x[i], w) / ||w|| );  pad slots get sentinel -2
// one wave per node
__global__ void score_kernel(const float* __restrict__ x, const float* __restrict__ w,
                             int n, int L, float* __restrict__ key, int* __restrict__ idx) {
  int gw   = (blockIdx.x * blockDim.x + threadIdx.x) >> 5;
  int lane = threadIdx.x & 31;
  if (gw >= L) return;
  if (gw < n) {
    float dot = 0.f, nrm = 0.f;
    #pragma unroll
    for (int t = 0; t < 4; t++) {
      int c = lane + 32 * t;
      float wv = w[c];
      dot += x[(size_t)gw * 128 + c] * wv;
      nrm += wv * wv;
    }
    dot = waveReduceSum(dot);
    nrm = waveReduceSum(nrm);
    if (lane == 0) { key[gw] = tanhf(dot * rsqrtf(nrm)); idx[gw] = gw; }
  } else {
    if (lane == 0) { key[gw] = -2.0f; idx[gw] = 0x40000000 + gw; }
  }
}

// one pass of a bitonic sort; final order: key descending, idx ascending on ties
__global__ void bitonic_step(float* __restrict__ key, int* __restrict__ idx,
                             int L, int boxw, int stride) {
  int i = blockIdx.x * blockDim.x + threadIdx.x;
  if (i >= L) return;
  int j = i ^ stride;
  if (j <= i) return;
  float ki = key[i], kj = key[j];
  int   ii = idx[i], ij = idx[j];
  bool iFirst = (ki > kj) || (ki == kj && ii < ij); // i ranks before j
  bool desc   = ((i & boxw) == 0);
  bool doswap = desc ? (!iFirst) : iFirst;
  if (doswap) { key[i] = kj; key[j] = ki; idx[i] = ij; idx[j] = ii; }
}

__global__ void fill_i32(int* __restrict__ p, int count, int val) {
  int i = blockIdx.x * blockDim.x + threadIdx.x;
  if (i < count) p[i] = val;
}

__global__ void scatter_perm(const int* __restrict__ sidx, int* __restrict__ perm,
                             int* __restrict__ nmap, int k) {
  int i = blockIdx.x * blockDim.x + threadIdx.x;
  if (i < k) { int p = sidx[i]; perm[i] = p; nmap[p] = i; }
}

// xp[i] = xold[perm[i]] * val[i] ; dummy row k = 0   (block = 128 thr, grid = k+1)
__global__ void gather_pool(const float* __restrict__ xold, const int* __restrict__ sidx,
                            const float* __restrict__ skey, float* __restrict__ xp, int k) {
  int row = blockIdx.x, c = threadIdx.x;
  float v = 0.f;
  if (row < k) v = xold[(size_t)sidx[row] * 128 + c] * skey[row];
  xp[(size_t)row * 128 + c] = v;
}

__global__ void remap_edges(const int* __restrict__ s, const int* __restrict__ r,
                            const int* __restrict__ nmap, int* __restrict__ so,
                            int* __restrict__ ro, int E, int k) {
  int i = blockIdx.x * blockDim.x + threadIdx.x;
  if (i >= E) return;
  int a = nmap[s[i]], b = nmap[r[i]];
  if (a == k || b == k) { a = k; b = k; }
  so[i] = a; ro[i] = b;
}

__global__ void zero_f32(float* __restrict__ p, size_t count) {
  size_t i = (size_t)blockIdx.x * blockDim.x + threadIdx.x;
  if (i < count) p[i] = 0.f;
}

// one wave per directed edge; lanes stride channels; skip dummy receivers
__global__ void scatter_add(const int* __restrict__ s, const int* __restrict__ r,
                            const float* __restrict__ xp, float* __restrict__ agg,
                            int E, int C, int dummy) {
  int gw   = (blockIdx.x * blockDim.x + threadIdx.x) >> 5;
  int lane = threadIdx.x & 31;
  if (gw >= 2 * E) return;
  int snd, rcv;
  if (gw < E) { snd = s[gw]; rcv = r[gw]; }
  else        { snd = r[gw - E]; rcv = s[gw - E]; }
  if (rcv == dummy) return;
  const float* src = xp  + (size_t)snd * C;
  float*       dst = agg + (size_t)rcv * C;
  for (int c = lane; c < C; c += 32) atomicAdd(&dst[c], src[c]);
}

// cat[row][0:128] = res[row] (row<n), rest 0   (block=256, grid=n+1)
__global__ void up_prepare(const float* __restrict__ res, float* __restrict__ cat, int n) {
  int row = blockIdx.x, c = threadIdx.x;
  float v = 0.f;
  if (row < n && c < 128) v = res[(size_t)row * 128 + c];
  cat[(size_t)row * 256 + c] = v;
}

// cat[perm[row]][128 + c] = xc[row][c]  (block=128, grid=k)
__global__ void up_scatter(const float* __restrict__ xc, const int* __restrict__ perm,
                           float* __restrict__ cat, int k) {
  int row = blockIdx.x, c = threadIdx.x;
  cat[(size_t)perm[row] * 256 + 128 + c] = xc[(size_t)row * 128 + c];
}

// ---------------------------------------------------------------------------
// out[M x 128] = act( A[M x KDIM] @ W[KDIM x 128] + b )   via f32 WMMA 16x16x4
// block = 256 threads (8 waves); block per 16-row tile; wave w -> cols 16w..16w+15
template<int KDIM, bool GELU_ACT>
__global__ __launch_bounds__(256)
void gemm_bias_act(const float* __restrict__ A, const float* __restrict__ W,
                   const float* __restrict__ bias, float* __restrict__ out, int M) {
  __shared__ float As[16][KDIM + 4];
  const int tid  = threadIdx.x;
  const int row0 = blockIdx.x * 16;

  for (int idx = tid; idx < 16 * KDIM; idx += 256) {
    int rr = idx / KDIM, cc = idx - rr * KDIM;
    int gr = row0 + rr;
    As[rr][cc] = (gr < M) ? A[(size_t)gr * KDIM + cc] : 0.f;
  }
  __syncthreads();

  const int wave = tid >> 5;
  const int lane = tid & 31;
  const int lm   = lane & 15;
  const int hi   = lane >> 4;            // 0: K pair {0,1}, 1: K pair {2,3}
  const int col  = wave * 16 + lm;       // output column (0..127)

  v8f acc;
  {
    float bv = bias[col];
    #pragma unroll
    for (int t = 0; t < 8; t++) acc[t] = bv;
  }

  #pragma unroll 8
  for (int k0 = 0; k0 < KDIM; k0 += 4) {
    int kk = k0 + hi * 2;
    v2f a, b;
    a.x = As[lm][kk];
    a.y = As[lm][kk + 1];
    b.x = W[(size_t)kk * 128 + col];
    b.y = W[(size_t)(kk + 1) * 128 + col];
    acc = __builtin_amdgcn_wmma_f32_16x16x4_f32(
        /*neg_a=*/false, a, /*neg_b=*/false, b,
        /*c_mod=*/(short)0, acc, /*reuse_a=*/false, /*reuse_b=*/false);
  }

  const int rowb = row0 + hi * 8;        // VGPR t -> row rowb + t
  #pragma unroll
  for (int t = 0; t < 8; t++) {
    int rr = rowb + t;
    if (rr < M) {
      float v = acc[t];
      if (GELU_ACT) v = 0.5f * v * (1.0f + erff(v * 0.70710678118654752f));
      out[(size_t)rr * 128 + col] = v;
    }
  }
}

// LayerNorm over 128 channels; one wave per row
__global__ void layernorm_kernel(const float* __restrict__ h, const float* __restrict__ g,
                                 const float* __restrict__ beta, float* __restrict__ out, int n) {
  int gw   = (blockIdx.x * blockDim.x + threadIdx.x) >> 5;
  int lane = threadIdx.x & 31;
  if (gw >= n) return;
  const float* hr = h + (size_t)gw * 128;
  float v[4], s = 0.f;
  #pragma unroll
  for (int t = 0; t < 4; t++) { v[t] = hr[lane + 32 * t]; s += v[t]; }
  s = waveReduceSum(s);
  float mu = s * (1.f / 128.f);
  float q = 0.f;
  #pragma unroll
  for (int t = 0; t < 4; t++) { float d = v[t] - mu; q += d * d; }
  q = waveReduceSum(q);
  float inv = rsqrtf(q * (1.f / 128.f) + 1e-5f);
  float* orow = out + (size_t)gw * 128;
  #pragma unroll
  for (int t = 0; t < 4; t++) {
    int c = lane + 32 * t;
    orow[c] = g[c] * (v[t] - mu) * inv + beta[c];
  }
}

// ---------------------------------------------------------------------------
static inline int next_pow2(int v) { int p = 1; while (p < v) p <<= 1; return p; }

extern "C" void kernel_launch(void* const* d_in, const int* in_sizes, int n_in,
                              void* d_out, int out_size, void* d_ws, size_t ws_size,
                              hipStream_t stream) {
  const float* x0    = (const float*)d_in[0];
  const int*   snd0  = (const int*)d_in[1];
  const int*   rcv0  = (const int*)d_in[2];
  const float* poolw = (const float*)d_in[3];
  const float* dW1 = (const float*)d_in[4];
  const float* db1 = (const float*)d_in[5];
  const float* dW2 = (const float*)d_in[6];
  const float* db2 = (const float*)d_in[7];
  const float* dW3 = (const float*)d_in[8];
  const float* db3 = (const float*)d_in[9];
  const float* dg  = (const float*)d_in[10];
  const float* dbe = (const float*)d_in[11];
  const float* uW1 = (const float*)d_in[12];
  const float* ub1 = (const float*)d_in[13];
  const float* uW2 = (const float*)d_in[14];
  const float* ub2 = (const float*)d_in[15];
  const float* uW3 = (const float*)d_in[16];
  const float* ub3 = (const float*)d_in[17];
  const float* ug  = (const float*)d_in[18];
  const float* ube = (const float*)d_in[19];

  const int C = 128;
  const int N = in_sizes[0] / C;   // 50000
  const int E = in_sizes[1];       // 600000

  int nlv[4];
  nlv[0] = N;
  for (int i = 0; i < 3; i++) nlv[i + 1] = (nlv[i] + 1) / 2;   // 25000,12500,6250
  const int Lmax = next_pow2(N);

  // --- workspace carve ---
  char* base = (char*)d_ws;
  size_t off = 0;
  auto carve = [&](size_t bytes) -> char* {
    off = (off + 255) & ~(size_t)255;
    char* p = base + off;
    off += bytes;
    return p;
  };
  float* sortkey = (float*)carve(sizeof(float) * (size_t)Lmax);
  int*   sortidx = (int*)  carve(sizeof(int)   * (size_t)Lmax);
  int*   permv[3];
  for (int i = 0; i < 3; i++) permv[i] = (int*)carve(sizeof(int) * (size_t)nlv[i + 1]);
  int*   nmap = (int*)carve(sizeof(int) * (size_t)(N + 1));
  int *sE[3], *rE[3];
  for (int i = 0; i < 3; i++) {
    sE[i] = (int*)carve(sizeof(int) * (size_t)E);
    rE[i] = (int*)carve(sizeof(int) * (size_t)E);
  }
  float* xlev[3];
  for (int i = 0; i < 3; i++) xlev[i] = (float*)carve(sizeof(float) * (size_t)nlv[i + 1] * C);
  float* xp  = (float*)carve(sizeof(float) * (size_t)(N + 1) * 256);
  float* agg = (float*)carve(sizeof(float) * (size_t)(N + 1) * 256);
  float* t1  = (float*)carve(sizeof(float) * (size_t)N * C);
  float* t2  = (float*)carve(sizeof(float) * (size_t)N * C);
  float* upx = (float*)carve(sizeof(float) * (size_t)nlv[1] * C);
  (void)ws_size; (void)n_in; (void)out_size;

  const int sblocks = (int)(((size_t)2 * E * 32 + 255) / 256);

  // ================= DOWN PATH =================
  const float* xcur = x0;
  const int* ps = snd0;
  const int* pr = rcv0;
  for (int lvl = 0; lvl < 3; lvl++) {
    int n = nlv[lvl], k = nlv[lvl + 1];
    int L = next_pow2(n);

    score_kernel<<<(L * 32 + 255) / 256, 256, 0, stream>>>(
        xcur, poolw + (size_t)lvl * C, n, L, sortkey, sortidx);
    for (int bw = 2; bw <= L; bw <<= 1)
      for (int st = bw >> 1; st > 0; st >>= 1)
        bitonic_step<<<(L + 255) / 256, 256, 0, stream>>>(sortkey, sortidx, L, bw, st);

    fill_i32<<<(n + 1 + 255) / 256, 256, 0, stream>>>(nmap, n + 1, k);
    scatter_perm<<<(k + 255) / 256, 256, 0, stream>>>(sortidx, permv[lvl], nmap, k);
    gather_pool<<<k + 1, 128, 0, stream>>>(xcur, sortidx, sortkey, xp, k);
    remap_edges<<<(E + 255) / 256, 256, 0, stream>>>(ps, pr, nmap, sE[lvl], rE[lvl], E, k);

    size_t aggc = (size_t)(k + 1) * C;
    zero_f32<<<(int)((aggc + 255) / 256), 256, 0, stream>>>(agg, aggc);
    scatter_add<<<sblocks, 256, 0, stream>>>(sE[lvl], rE[lvl], xp, agg, E, C, k);

    gemm_bias_act<128, true ><<<(k + 15) / 16, 256, 0, stream>>>(
        agg, dW1 + (size_t)lvl * C * C, db1 + (size_t)lvl * C, t1, k);
    gemm_bias_act<128, true ><<<(k + 15) / 16, 256, 0, stream>>>(
        t1, dW2 + (size_t)lvl * C * C, db2 + (size_t)lvl * C, t2, k);
    gemm_bias_act<128, false><<<(k + 15) / 16, 256, 0, stream>>>(
        t2, dW3 + (size_t)lvl * C * C, db3 + (size_t)lvl * C, t1, k);
    layernorm_kernel<<<(k * 32 + 255) / 256, 256, 0, stream>>>(
        t1, dg + (size_t)lvl * C, dbe + (size_t)lvl * C, xlev[lvl], k);

    xcur = xlev[lvl];
    ps = sE[lvl];
    pr = rE[lvl];
  }

  // ================= UP PATH =================
  const float* xu = xlev[2];   // 6250 x 128
  for (int i = 0; i < 3; i++) {
    int j  = 2 - i;
    int nj = nlv[j];           // rows of res
    int kj = nlv[j + 1];       // rows of xu
    const float* res = (j == 0) ? x0   : xlev[j - 1];
    const int*   es  = (j == 0) ? snd0 : sE[j - 1];
    const int*   er  = (j == 0) ? rcv0 : rE[j - 1];
    const int*   pj  = permv[j];

    up_prepare<<<nj + 1, 256, 0, stream>>>(res, xp, nj);
    up_scatter<<<kj, 128, 0, stream>>>(xu, pj, xp, kj);

    size_t aggc = (size_t)(nj + 1) * 256;
    zero_f32<<<(int)((aggc + 255) / 256), 256, 0, stream>>>(agg, aggc);
    scatter_add<<<sblocks, 256, 0, stream>>>(es, er, xp, agg, E, 256, nj);

    float* dest = (i == 2) ? (float*)d_out : upx;
    gemm_bias_act<256, true ><<<(nj + 15) / 16, 256, 0, stream>>>(
        agg, uW1 + (size_t)i * 256 * C, ub1 + (size_t)i * C, t1, nj);
    gemm_bias_act<128, true ><<<(nj + 15) / 16, 256, 0, stream>>>(
        t1, uW2 + (size_t)i * C * C, ub2 + (size_t)i * C, t2, nj);
    gemm_bias_act<128, false><<<(nj + 15) / 16, 256, 0, stream>>>(
        t2, uW3 + (size_t)i * C * C, ub3 + (size_t)i * C, t1, nj);
    layernorm_kernel<<<(nj * 32 + 255) / 256, 256, 0, stream>>>(
        t1, ug + (size_t)i * C, ube + (size_t)i * C, dest, nj);

    xu = dest;
  }
}